// PCA_21517786153586
// MI455X (gfx1250) — compile-verified
//
#include <hip/hip_runtime.h>

typedef __attribute__((ext_vector_type(2))) float v2f;
typedef __attribute__((ext_vector_type(8))) float v8f;
// align(4) 2-float vector: lets LLVM emit one (possibly misaligned) b64 load
// instead of two b32 loads; K dims are odd so 8B alignment is unprovable.
typedef float f32x2u __attribute__((ext_vector_type(2), aligned(4)));

#define N_VERTS   53215
#define N_ROWS    (3 * N_VERTS)      // 159645
#define EXP_DIM   29
#define SHP_DIM   199
#define HEIGHT_C  450.0f
#define ASPECT_XY (224.0f / 450.0f)

// ---------------------------------------------------------------------------
// Kernel A: vertices = u + W_shp * a_shp + W_exp * a_exp   (per-row GEMV)
// One wave32 = one 16-row tile, accumulated with V_WMMA_F32_16X16X4_F32.
// A-fragment layout (ISA 7.12.2, 32-bit A 16x4): lanes 0-15 hold K=k0,k0+1,
// lanes 16-31 hold K=k0+2,k0+3, for row M = lane&15.
// B-fragment: alpha in column N=0 only => nonzero only in lanes 0 and 16;
// alpha staged zero-padded in LDS so the hot loop needs no bounds checks.
// C/D layout: lane 0 holds D[0..7][0] in c[0..7]; lane 16 holds D[8..15][0].
// K-tail (the only place clamping is live) is peeled so the main loop can
// use vectorized b64 A-loads.
// ---------------------------------------------------------------------------
__global__ __launch_bounds__(256) void pca_gemv_wmma(
    const float* __restrict__ alpha_exp,
    const float* __restrict__ alpha_shp,
    const float* __restrict__ u_base,
    const float* __restrict__ w_exp,
    const float* __restrict__ w_shp,
    float* __restrict__ vtx)
{
    __shared__ float s_shp[200];  // SHP_DIM padded to multiple of 4, zero tail
    __shared__ float s_exp[32];   // EXP_DIM padded to multiple of 4, zero tail

    const int tid = threadIdx.x;
    if (tid < 200) s_shp[tid] = (tid < SHP_DIM) ? alpha_shp[tid] : 0.0f;
    if (tid < 32)  s_exp[tid] = (tid < EXP_DIM) ? alpha_exp[tid] : 0.0f;
    __syncthreads();

    const int lane  = tid & 31;
    const int wave  = tid >> 5;
    const int base  = (blockIdx.x * 8 + wave) * 16;   // tile's first row
    if (base >= N_ROWS) return;                        // uniform per wave

    const int m     = lane & 15;                       // row within tile
    const int khalf = (lane >> 4) << 1;                // 0 (lo half) or 2 (hi)

    int row = base + m;
    if (row > N_ROWS - 1) row = N_ROWS - 1;            // clamp, keep EXEC full

    v8f c = {};

    // ---- shp contribution: K = 199 -> 49 full steps + 1 peeled tail ----
    {
        const float* rp = w_shp + (long)row * SHP_DIM;
        for (int k0 = 0; k0 < 196; k0 += 4) {
            const int ka = k0 + khalf;
            const f32x2u av = *(const f32x2u*)(rp + ka);   // one b64 load
            v2f a; a.x = av.x; a.y = av.y;
            v2f b;
            b.x = (m == 0) ? s_shp[ka]     : 0.0f;         // col N=0 only
            b.y = (m == 0) ? s_shp[ka + 1] : 0.0f;
            c = __builtin_amdgcn_wmma_f32_16x16x4_f32(
                    false, a, false, b, (short)0, c, false, false);
        }
        // tail: k0 = 196, only element 199 is out of range -> clamp A, B pad=0
        {
            const int ka  = 196 + khalf;
            const int ka1 = (ka + 1 < SHP_DIM) ? ka + 1 : SHP_DIM - 1;
            v2f a; a.x = rp[ka]; a.y = rp[ka1];
            v2f b;
            b.x = (m == 0) ? s_shp[ka]     : 0.0f;
            b.y = (m == 0) ? s_shp[ka + 1] : 0.0f;         // s_shp[199] == 0
            c = __builtin_amdgcn_wmma_f32_16x16x4_f32(
                    false, a, false, b, (short)0, c, false, false);
        }
    }

    // ---- exp contribution: K = 29 -> 7 full steps + 1 peeled tail ----
    {
        const float* rp = w_exp + (long)row * EXP_DIM;
        for (int k0 = 0; k0 < 28; k0 += 4) {
            const int ka = k0 + khalf;
            const f32x2u av = *(const f32x2u*)(rp + ka);   // one b64 load
            v2f a; a.x = av.x; a.y = av.y;
            v2f b;
            b.x = (m == 0) ? s_exp[ka]     : 0.0f;
            b.y = (m == 0) ? s_exp[ka + 1] : 0.0f;
            c = __builtin_amdgcn_wmma_f32_16x16x4_f32(
                    false, a, false, b, (short)0, c, false, false);
        }
        // tail: k0 = 28, elements 29..31 out of range -> clamp A, B pad=0
        {
            const int ka  = 28 + khalf;
            const int ka0 = (ka     < EXP_DIM) ? ka     : EXP_DIM - 1;
            const int ka1 = (ka + 1 < EXP_DIM) ? ka + 1 : EXP_DIM - 1;
            v2f a; a.x = rp[ka0]; a.y = rp[ka1];
            v2f b;
            b.x = (m == 0) ? s_exp[ka]     : 0.0f;         // s_exp[29..31]==0
            b.y = (m == 0) ? s_exp[ka + 1] : 0.0f;
            c = __builtin_amdgcn_wmma_f32_16x16x4_f32(
                    false, a, false, b, (short)0, c, false, false);
        }
    }

    // ---- epilogue: lanes 0 and 16 hold column 0 of D ----
    if (m == 0) {
        const int rbase = base + (khalf << 2);         // lane0: +0, lane16: +8
        #pragma unroll
        for (int r = 0; r < 8; ++r) {
            const int rr = rbase + r;
            if (rr < N_ROWS) vtx[rr] = c[r] + u_base[rr];
        }
    }
}

// ---------------------------------------------------------------------------
// Kernel B: affine transform of (x,y,z,1) by T = H @ M @ S @ T0, times ASPECT.
//   s = p3+p7+p11 ; t = (p3, p7, 0)
//   out0 = ( s*(p0 x + p1 y + p2 z) + p3          ) * 224/450
//   out1 = (-s*(p4 x + p5 y + p6 z) - p7 + 450.0  ) * 224/450
//   out2 =   s*(p8 x + p9 y + p10 z)
// ---------------------------------------------------------------------------
__global__ __launch_bounds__(256) void pca_transform(
    const float* __restrict__ pose,
    const float* __restrict__ vtx,
    float* __restrict__ out)
{
    const int i = blockIdx.x * blockDim.x + threadIdx.x;
    if (i >= N_VERTS) return;

    float p[12];
    #pragma unroll
    for (int j = 0; j < 12; ++j) p[j] = pose[j];
    const float s = p[3] + p[7] + p[11];

    const float x = vtx[3 * i + 0];
    const float y = vtx[3 * i + 1];
    const float z = vtx[3 * i + 2];

    const float o0 = ( s * (p[0] * x + p[1] * y + p[2]  * z) + p[3]) * ASPECT_XY;
    const float o1 = (-s * (p[4] * x + p[5] * y + p[6]  * z) - p[7] + HEIGHT_C) * ASPECT_XY;
    const float o2 =   s * (p[8] * x + p[9] * y + p[10] * z);

    out[3 * i + 0] = o0;
    out[3 * i + 1] = o1;
    out[3 * i + 2] = o2;
}

extern "C" void kernel_launch(void* const* d_in, const int* in_sizes, int n_in,
                              void* d_out, int out_size, void* d_ws, size_t ws_size,
                              hipStream_t stream) {
    const float* pose  = (const float*)d_in[0];  // (1,12)
    const float* a_exp = (const float*)d_in[1];  // (1,29)
    const float* a_shp = (const float*)d_in[2];  // (1,199)
    const float* u     = (const float*)d_in[3];  // (N_ROWS,1)
    const float* w_exp = (const float*)d_in[4];  // (N_ROWS,29)
    const float* w_shp = (const float*)d_in[5];  // (N_ROWS,199)
    float* out = (float*)d_out;                  // (1,N_VERTS,3)
    float* vtx = (float*)d_ws;                   // N_ROWS floats scratch

    const int tiles   = (N_ROWS + 15) / 16;      // 9978 row tiles
    const int blocksA = (tiles + 7) / 8;         // 8 waves / block
    pca_gemv_wmma<<<blocksA, 256, 0, stream>>>(a_exp, a_shp, u, w_exp, w_shp, vtx);

    const int blocksB = (N_VERTS + 255) / 256;
    pca_transform<<<blocksB, 256, 0, stream>>>(pose, vtx, out);
}